// HashGridEncoder_89215060672776
// MI455X (gfx1250) — compile-verified
//
#include <hip/hip_runtime.h>
#include <hip/hip_bf16.h>
#include <stdint.h>

// ---------------------------------------------------------------------------
// Instant-NGP hash-grid forward encoder for gfx1250 (MI455X).
// Gather-bound: 64 MB table is fully L2-resident (192 MB L2); we maximize
// outstanding global_load_b64 gathers per wave, stream the 128 MB output with
// non-temporal stores so it does not evict the table from L2, and stage the
// input points through LDS with the CDNA5 async global->LDS path (ASYNCcnt).
// ---------------------------------------------------------------------------

#define HG_LEVELS     16
#define HG_LOG2T      19
#define HG_TSIZE      (1u << HG_LOG2T)
#define HG_TMASK      (HG_TSIZE - 1u)
#define HG_NDENSE     5          // levels 0..4: (res+1)^3 <= 2^19
#define HG_BLOCK      256        // 8 wave32 waves

typedef __attribute__((ext_vector_type(4))) float v4f;      // native vec for NT store
typedef __attribute__((address_space(1))) int    as1_int;   // global (__device__)
typedef __attribute__((address_space(3))) int    as3_int;   // LDS (__shared__)

// floor(16 * (2048/16)^(l/15)) for l = 0..15 (matches the numpy reference)
__device__ constexpr uint32_t kRes[HG_LEVELS] = {
    16, 22, 30, 42, 58, 80, 111, 153, 212, 294, 406, 561, 776, 1072, 1482, 2048};

#if defined(__has_builtin)
#if __has_builtin(__builtin_amdgcn_global_load_async_to_lds_b32) && \
    __has_builtin(__builtin_amdgcn_s_wait_asynccnt)
#define HG_HAVE_ASYNC_LDS 1
#endif
#endif
#ifndef HG_HAVE_ASYNC_LDS
#define HG_HAVE_ASYNC_LDS 0
#endif

__global__ __launch_bounds__(HG_BLOCK) void hashgrid_fwd_kernel(
    const float* __restrict__ x,      // [n, 3]
    const float* __restrict__ table,  // [16, 2^19, 2]
    float* __restrict__ out,          // [n, 32]
    int n)
{
    __shared__ float sx[HG_BLOCK * 3];

    const int t    = threadIdx.x;
    const int base = blockIdx.x * HG_BLOCK;
    const int todo = (n - base < HG_BLOCK ? n - base : HG_BLOCK) * 3;

    // ---- Stage this block's x slice into LDS (coalesced b32 stream). ----
    const float* gx = x + (size_t)base * 3;
#if HG_HAVE_ASYNC_LDS
    #pragma unroll
    for (int j = 0; j < 3; ++j) {
        const int i = t + HG_BLOCK * j;
        if (i < todo) {
            __builtin_amdgcn_global_load_async_to_lds_b32(
                (as1_int*)(gx + i),
                (as3_int*)(sx + i),
                /*offset=*/0, /*cpol=*/0);
        }
    }
    __builtin_amdgcn_s_wait_asynccnt(0);   // s_wait_asynccnt 0
    __syncthreads();
#else
    #pragma unroll
    for (int j = 0; j < 3; ++j) {
        const int i = t + HG_BLOCK * j;
        if (i < todo) sx[i] = gx[i];
    }
    __syncthreads();
#endif

    const int p = base + t;
    if (p >= n) return;

    // xc = clip(x, 0, 1)
    float xc[3];
    #pragma unroll
    for (int d = 0; d < 3; ++d)
        xc[d] = fminf(fmaxf(sx[t * 3 + d], 0.0f), 1.0f);

    float o[2 * HG_LEVELS];  // register-resident accumulators

    #pragma unroll
    for (int l = 0; l < HG_LEVELS; ++l) {
        const uint32_t res  = kRes[l];
        const float    resf = (float)res;

        // pos = xc*res; pos0 = min(floor(pos), res-1); frac = pos - pos0
        float    fr[3];
        uint32_t ci[3];
        #pragma unroll
        for (int d = 0; d < 3; ++d) {
            const float pos = xc[d] * resf;
            float p0 = floorf(pos);
            p0 = fminf(p0, resf - 1.0f);
            fr[d] = pos - p0;
            ci[d] = (uint32_t)p0;
        }

        const float* tl = table + (size_t)l * (size_t)HG_TSIZE * 2u;

        float acc0 = 0.0f, acc1 = 0.0f;
        #pragma unroll
        for (int c = 0; c < 8; ++c) {
            const uint32_t cx = ci[0] + ( c       & 1u);
            const uint32_t cy = ci[1] + ((c >> 1) & 1u);
            const uint32_t cz = ci[2] + ((c >> 2) & 1u);

            uint32_t idx;
            if (l < HG_NDENSE) {
                const uint32_t r1 = res + 1u;
                idx = cx + r1 * (cy + r1 * cz);           // dense tiling
            } else {
                idx = cx ^ (cy * 2654435761u) ^ (cz * 805459861u);
                idx &= HG_TMASK;                           // spatial hash
            }

            // trilinear weight (dim order matches jnp.prod over axis -1)
            const float wx = (c & 1) ? fr[0] : 1.0f - fr[0];
            const float wy = (c & 2) ? fr[1] : 1.0f - fr[1];
            const float wz = (c & 4) ? fr[2] : 1.0f - fr[2];
            const float w  = wx * wy * wz;

            // 32-bit byte offset -> saddr + voffset global_load_b64 form
            const uint32_t boff = idx * 8u;
            const float2 f = *(const float2*)((const char*)tl + boff);
            acc0 = fmaf(w, f.x, acc0);
            acc1 = fmaf(w, f.y, acc1);
        }
        o[2 * l + 0] = acc0;
        o[2 * l + 1] = acc1;
    }

    // ---- Streamed output: non-temporal b128 stores keep the table in L2 ----
    v4f* op = (v4f*)(out + (size_t)p * (2 * HG_LEVELS));
    #pragma unroll
    for (int k = 0; k < 8; ++k) {
        v4f v;
        v.x = o[4 * k + 0];
        v.y = o[4 * k + 1];
        v.z = o[4 * k + 2];
        v.w = o[4 * k + 3];
        __builtin_nontemporal_store(v, op + k);
    }
}

extern "C" void kernel_launch(void* const* d_in, const int* in_sizes, int n_in,
                              void* d_out, int out_size, void* d_ws, size_t ws_size,
                              hipStream_t stream) {
    const float* x     = (const float*)d_in[0];   // [N, 3] f32
    const float* table = (const float*)d_in[1];   // [16, 2^19, 2] f32
    float*       out   = (float*)d_out;           // [N, 32] f32

    const int n      = in_sizes[0] / 3;
    const int blocks = (n + HG_BLOCK - 1) / HG_BLOCK;

    hipLaunchKernelGGL(hashgrid_fwd_kernel, dim3(blocks), dim3(HG_BLOCK), 0, stream,
                       x, table, out, n);
}